// Attention_25108378812747
// MI455X (gfx1250) — compile-verified
//
#include <hip/hip_runtime.h>
#include <hip/hip_bf16.h>
#include <cmath>
#include <cstdint>

#define B_   256
#define L_   196
#define ENC  2048
#define DEC  512
#define HID  512

#define KC    256            // K-chunk (floats) staged per TDM tile
#define ROWD  260            // LDS dwords per row: 256 + 4 pad (1 dword per 64) -> 4-bank shift/row
#define BUFD  (16 * ROWD)    // dwords per buffer (4160)
#define NMT   13             // ceil(196/16) M-tiles
#define NCH   (ENC / KC)     // 8 K-chunks
#define NG    (NMT * NCH)    // 104 chunks total

typedef float v2f __attribute__((ext_vector_type(2)));
typedef float v8f __attribute__((ext_vector_type(8)));
typedef unsigned int u32x4 __attribute__((ext_vector_type(4)));
typedef int i32x4 __attribute__((ext_vector_type(4)));
typedef int i32x8 __attribute__((ext_vector_type(8)));

#ifndef __has_builtin
#define __has_builtin(x) 0
#endif
#if defined(__gfx1250__) && __has_builtin(__builtin_amdgcn_tensor_load_to_lds)
#define USE_TDM 1
#else
#define USE_TDM 0
#endif

// ---------------------------------------------------------------------------
// Kernel 1: hb[b][h] = hidden[b,:]·Wh[h,:] + bh[h] + bf[h]   (tiny: 0.13 GF)
// ---------------------------------------------------------------------------
__global__ __launch_bounds__(512) void hb_kernel(const float* __restrict__ hidden,
                                                 const float* __restrict__ Wh,
                                                 const float* __restrict__ bh,
                                                 const float* __restrict__ bf,
                                                 float* __restrict__ hb) {
    __shared__ float sh[DEC];
    const int b = blockIdx.x;
    const int h = threadIdx.x;              // 512 threads == HID
    sh[h] = hidden[b * DEC + h];
    __syncthreads();
    const float4* wr = (const float4*)(Wh + (size_t)h * DEC);
    float s = 0.f;
    #pragma unroll 8
    for (int e4 = 0; e4 < DEC / 4; ++e4) {
        float4 w = wr[e4];
        int e = e4 * 4;
        s += w.x * sh[e] + w.y * sh[e + 1] + w.z * sh[e + 2] + w.w * sh[e + 3];
    }
    hb[b * HID + h] = s + bh[h] + bf[h];
}

#if USE_TDM
// ---------------------------------------------------------------------------
// Issue one TDM tile load: 16 rows x KC floats of `feature[b]` -> LDS buffer.
// D# packing per CDNA5 ISA sec 8.3-8.6:
//   group0: count=1 | lds_addr | global_addr(57b) | type=2
//   group1: data_size=4B, pad_enable, pad_interval=64dw, pad_amount=1dw,
//           tensor_dim0/1 (relative to tile start: OOB rows read as zero),
//           tile_dim0=KC, tile_dim1=16, tensor_dim0_stride=ENC
//   groups 2/3 + trailing group: zero (2-D tensor)
// This toolchain exposes the 6-arg builtin:
//   (u32x4 g0, i32x8 g1, i32x4 g2, i32x4 g3, i32x8 g4, i32 cpol)
// ---------------------------------------------------------------------------
__device__ __forceinline__ void tdm_issue(const float* gtile, unsigned lds_byte,
                                          int td0, int td1) {
    unsigned long long ga = (unsigned long long)(uintptr_t)gtile;
    u32x4 g0;
    g0[0] = 1u;                                        // count=1, gather off
    g0[1] = lds_byte;                                  // LDS byte address
    g0[2] = (unsigned)ga;                              // global_addr[31:0]
    g0[3] = (unsigned)(ga >> 32) | 0x80000000u;        // global_addr[56:32] | type=2

    i32x8 g1;
    g1[0] = (2 << 16) | (1 << 20) | (5 << 22);         // data_size=4B | pad_en | interval=64dw | amount=1dw
    g1[1] = (td0 & 0xFFFF) << 16;                      // tensor_dim0[15:0]
    g1[2] = ((td0 >> 16) & 0xFFFF) | ((td1 & 0xFFFF) << 16);   // td0 hi | tensor_dim1 lo
    g1[3] = ((td1 >> 16) & 0xFFFF) | (KC << 16);       // td1 hi | tile_dim0 = KC
    g1[4] = 16;                                        // tile_dim1 = 16 rows, tile_dim2 = 0
    g1[5] = ENC;                                       // tensor_dim0_stride[31:0] = 2048
    g1[6] = 0;                                         // stride0 hi | stride1 lo
    g1[7] = 0;                                         // stride1 hi

    i32x4 gz4;
    gz4[0] = 0; gz4[1] = 0; gz4[2] = 0; gz4[3] = 0;
    i32x8 gz8;
    gz8[0] = 0; gz8[1] = 0; gz8[2] = 0; gz8[3] = 0;
    gz8[4] = 0; gz8[5] = 0; gz8[6] = 0; gz8[7] = 0;

    __builtin_amdgcn_tensor_load_to_lds(g0, g1, gz4, gz4, gz8, 0);
}
#endif

// ---------------------------------------------------------------------------
// Kernel 2: fully fused per-batch-row attention. One workgroup per b.
//   score[l] = We · relu(Wf·feature[b,l] + hb[b]) + be   (fp32 WMMA GEMM)
//   weight   = softmax(score)                            (in LDS)
//   output[b]= weightᵀ · feature[b]                      (L2-hot re-read)
// ---------------------------------------------------------------------------
__global__ __launch_bounds__(512) void attn_kernel(const float* __restrict__ feature,
                                                   const float* __restrict__ Wf,
                                                   const float* __restrict__ We,
                                                   const float* __restrict__ be,
                                                   const float* __restrict__ hb,
                                                   float* __restrict__ out_out,  // [B, ENC]
                                                   float* __restrict__ out_w) {  // [B, L]
    __shared__ float sA[2 * BUFD];   // double-buffered tile, 2 x 16.25KB
    __shared__ float sScore[224];    // scores (padded 13*16=208 -> 224)
    __shared__ float sRed[16];       // per-wave reduction scratch

    const int b     = blockIdx.x;
    const int tid   = threadIdx.x;
    const int lane  = tid & 31;
    const int wave  = tid >> 5;       // 0..15
    const int row16 = lane & 15;      // A-frag row (M) / B-frag row (N) / C col (N)
    const int khalf = lane >> 4;      // 0/1: which K-pair of the fragment

    const float be0 = be[0];
    for (int t = tid; t < 224; t += 512) sScore[t] = be0;   // score starts at bias

    const float* fb = feature + (size_t)b * (L_ * ENC);
    const int h0 = wave * 32 + row16;   // wave owns N-tiles {2w, 2w+1}
    const int h1 = h0 + 16;

#if USE_TDM
    const unsigned ldsBase = (unsigned)(uintptr_t)&sA[0];
    if (wave == 0) {
        tdm_issue(fb, ldsBase, ENC, L_);            // chunk 0: l0=0, e0=0
    }
#endif
    __syncthreads();

    for (int mt = 0; mt < NMT; ++mt) {
        v8f acc0 = {};
        v8f acc1 = {};

        for (int kc = 0; kc < NCH; ++kc) {
            const int cur = mt * NCH + kc;
            const int buf = cur & 1;
            const int e0  = kc * KC;

#if USE_TDM
            if (wave == 0) {
                const int nxt = cur + 1;
                if (nxt < NG) {
                    const int nl0 = (nxt >> 3) * 16;
                    const int ne0 = (nxt & 7) * KC;
                    tdm_issue(fb + (size_t)nl0 * ENC + ne0,
                              ldsBase + (unsigned)((nxt & 1) * (BUFD * 4)),
                              ENC - ne0, L_ - nl0);
                    __builtin_amdgcn_s_wait_tensorcnt(1);  // in-order => chunk `cur` landed
                } else {
                    __builtin_amdgcn_s_wait_tensorcnt(0);
                }
            }
            __syncthreads();
#else
            __syncthreads();   // previous use of this buffer fully consumed
            {
                const int l0 = mt * 16;
                #pragma unroll
                for (int j = 0; j < 2; ++j) {
                    int f = tid + j * 512;          // 0..1023 float4 slots
                    int r = f >> 6;                 // 64 float4 per row
                    int c = (f & 63) * 4;           // dword column in row
                    float4 v = make_float4(0.f, 0.f, 0.f, 0.f);
                    int l = l0 + r;
                    if (l < L_) v = *(const float4*)(fb + (size_t)l * ENC + e0 + c);
                    float* dst = &sA[buf * BUFD + r * ROWD + c + (c >> 6)];
                    dst[0] = v.x; dst[1] = v.y; dst[2] = v.z; dst[3] = v.w;
                }
            }
            __syncthreads();
#endif

            const float* abase = &sA[buf * BUFD + row16 * ROWD + 2 * khalf];
            const float* bp0   = Wf + (size_t)h0 * ENC + e0 + 2 * khalf;
            const float* bp1   = Wf + (size_t)h1 * ENC + e0 + 2 * khalf;
            #pragma unroll 8
            for (int kk = 0; kk < KC; kk += 4) {
                v2f a   = *(const v2f*)(abase + kk + (kk >> 6));  // in-row pad shift
                v2f bv0 = *(const v2f*)(bp0 + kk);
                v2f bv1 = *(const v2f*)(bp1 + kk);
                acc0 = __builtin_amdgcn_wmma_f32_16x16x4_f32(
                    false, a, false, bv0, (short)0, acc0, false, false);
                acc1 = __builtin_amdgcn_wmma_f32_16x16x4_f32(
                    false, a, false, bv1, (short)0, acc1, false, false);
            }
            __syncthreads();   // buffer consumed; wave0 may overwrite it next round
        }

        // epilogue: v = relu(acc + hb[h]) * We[h]; reduce over N into score[m]
        const int   l0  = mt * 16;
        const float hb0 = hb[b * HID + h0], hb1 = hb[b * HID + h1];
        const float we0 = We[h0],           we1 = We[h1];
        float p[8];
        #pragma unroll
        for (int i = 0; i < 8; ++i) {
            float v0 = fmaxf(acc0[i] + hb0, 0.f) * we0;
            float v1 = fmaxf(acc1[i] + hb1, 0.f) * we1;
            p[i] = v0 + v1;     // C layout: N=row16, M = i + 8*khalf
        }
        #pragma unroll
        for (int i = 0; i < 8; ++i) {   // sum over the 16 lanes of each half
            p[i] += __shfl_xor(p[i], 1);
            p[i] += __shfl_xor(p[i], 2);
            p[i] += __shfl_xor(p[i], 4);
            p[i] += __shfl_xor(p[i], 8);
        }
        if (row16 == 0) {
            #pragma unroll
            for (int i = 0; i < 8; ++i)
                atomicAdd(&sScore[l0 + khalf * 8 + i], p[i]);   // ds_add_f32
        }
    }
    __syncthreads();

    // ---- softmax over sScore[0..195] ----
    float x = (tid < L_) ? sScore[tid] : -__builtin_inff();
    float m = x;
    m = fmaxf(m, __shfl_xor(m, 16));
    m = fmaxf(m, __shfl_xor(m, 8));
    m = fmaxf(m, __shfl_xor(m, 4));
    m = fmaxf(m, __shfl_xor(m, 2));
    m = fmaxf(m, __shfl_xor(m, 1));
    if (lane == 0) sRed[wave] = m;
    __syncthreads();
    float mx = sRed[0];
    #pragma unroll
    for (int wv = 1; wv < 16; ++wv) mx = fmaxf(mx, sRed[wv]);
    float ex = (tid < L_) ? __expf(x - mx) : 0.f;
    __syncthreads();                       // done reading sRed (max phase)
    float s = ex;
    s += __shfl_xor(s, 16);
    s += __shfl_xor(s, 8);
    s += __shfl_xor(s, 4);
    s += __shfl_xor(s, 2);
    s += __shfl_xor(s, 1);
    if (lane == 0) sRed[wave] = s;
    __syncthreads();
    float tot = 0.f;
    #pragma unroll
    for (int wv = 0; wv < 16; ++wv) tot += sRed[wv];
    const float w = ex / tot;
    __syncthreads();                       // everyone read their sScore[tid] already
    if (tid < L_) {
        sScore[tid] = w;
        out_w[b * L_ + tid] = w;
    }
    __syncthreads();

    // ---- output[b,e] = sum_l w[l] * feature[b,l,e]  (feature[b] is L2-hot) ----
    float4 accv = make_float4(0.f, 0.f, 0.f, 0.f);
    const float* fcol = fb + tid * 4;      // 512 threads * float4 == 2048 cols
    #pragma unroll 4
    for (int l = 0; l < L_; ++l) {
        float wl = sScore[l];
        float4 v = *(const float4*)(fcol + (size_t)l * ENC);
        accv.x += wl * v.x;
        accv.y += wl * v.y;
        accv.z += wl * v.z;
        accv.w += wl * v.w;
    }
    *(float4*)(out_out + (size_t)b * ENC + tid * 4) = accv;
}

// ---------------------------------------------------------------------------
extern "C" void kernel_launch(void* const* d_in, const int* in_sizes, int n_in,
                              void* d_out, int out_size, void* d_ws, size_t ws_size,
                              hipStream_t stream) {
    const float* feature = (const float*)d_in[0];
    const float* hidden  = (const float*)d_in[1];
    const float* Wf      = (const float*)d_in[2];
    const float* bf      = (const float*)d_in[3];
    const float* Wh      = (const float*)d_in[4];
    const float* bh      = (const float*)d_in[5];
    const float* We      = (const float*)d_in[6];
    const float* be      = (const float*)d_in[7];

    float* out = (float*)d_out;                 // [B,ENC] output, then [B,L] weight
    float* hb  = (float*)d_ws;                  // [B,HID] = 512 KB scratch

    hb_kernel<<<B_, 512, 0, stream>>>(hidden, Wh, bh, bf, hb);
    attn_kernel<<<B_, 512, 0, stream>>>(feature, Wf, We, be, hb,
                                        out, out + (size_t)B_ * ENC);
}